// DynamicGraphConv_36026185678969
// MI455X (gfx1250) — compile-verified
//
#include <hip/hip_runtime.h>

#define N_NODES 100000
#define N_EDGES 1000000
#define DIM     64
#define OUTF    64
#define EPS_LN  1e-5f

typedef __attribute__((ext_vector_type(2))) float v2f;
typedef __attribute__((ext_vector_type(8))) float v8f;

// ---------------------------------------------------------------------------
// Kernel 1: zero the neighbor-sum accumulator in workspace (float4 stores).
// ---------------------------------------------------------------------------
__global__ void dgc_zero_kernel(float4* __restrict__ ws, int n4) {
  int i = blockIdx.x * blockDim.x + threadIdx.x;
  if (i < n4) ws[i] = make_float4(0.f, 0.f, 0.f, 0.f);
}

// ---------------------------------------------------------------------------
// Kernel 2: edge scatter. 16 threads per edge, each moves a float4 chunk of
// the 64-float source row and atomically accumulates into nsum[dst].
// x (25.6MB) and nsum (25.6MB) are both L2-resident on MI455X (192MB L2);
// this stage is bound by L2 atomic throughput -> we need the native
// global_atomic_add_f32 path, so use relaxed agent-scope HIP atomics.
// ---------------------------------------------------------------------------
__global__ void dgc_scatter_kernel(const float* __restrict__ x,
                                   const long long* __restrict__ ei,
                                   float* __restrict__ nsum) {
  unsigned tid = blockIdx.x * blockDim.x + threadIdx.x;
  unsigned e = tid >> 4;
  if (e >= N_EDGES) return;
  int c = (tid & 15) << 2;                     // float4 chunk within the row
  long long src = ei[e];                       // edge_index[0][e]
  long long dst = ei[(long long)N_EDGES + e];  // edge_index[1][e]
  float4 v = *(const float4*)(x + src * DIM + c);
  float* p = nsum + dst * DIM + c;
  __hip_atomic_fetch_add(p + 0, v.x, __ATOMIC_RELAXED, __HIP_MEMORY_SCOPE_AGENT);
  __hip_atomic_fetch_add(p + 1, v.y, __ATOMIC_RELAXED, __HIP_MEMORY_SCOPE_AGENT);
  __hip_atomic_fetch_add(p + 2, v.z, __ATOMIC_RELAXED, __HIP_MEMORY_SCOPE_AGENT);
  __hip_atomic_fetch_add(p + 3, v.w, __ATOMIC_RELAXED, __HIP_MEMORY_SCOPE_AGENT);
}

// ---------------------------------------------------------------------------
// Kernel 3: fused  h = [x | nsum] @ W + b  followed by LayerNorm(64).
// Block = 256 threads = 8 wave32 waves, covering 32 nodes.
//   wave w: row tile = (w>>2)*16, n tile = (w&3)*16  -> 2x4 tiling of 32x64.
// 32 x V_WMMA_F32_16X16X4_F32 per wave (K=128 in steps of 4), exact f32.
// W is repacked in LDS as interleaved row-pairs so each lane's B fragment
// {W[k][n], W[k+1][n]} is one aligned ds_load_b64.
// ---------------------------------------------------------------------------
#define WP_STRIDE 130   // floats per packed W row-pair (128 + 2 pad)

__global__ __launch_bounds__(256) void dgc_gemm_ln_kernel(
    const float* __restrict__ x, const float* __restrict__ nsum,
    const float* __restrict__ W, const float* __restrict__ bias,
    const float* __restrict__ gamma, const float* __restrict__ beta,
    float* __restrict__ out) {
  __shared__ float sW2[64 * WP_STRIDE];  // W row-pairs: [k>>1][2n + (k&1)]
  __shared__ float sA[32 * 130];         // A tiles [32 nodes][128], pad 130
  __shared__ float sH[32 * 68];          // pre-LN result [32][64], pad 68
  __shared__ float sStat[64];            // mu[32] | rsigma[32]

  const int t = threadIdx.x;
  const int nodeBase = blockIdx.x * 32;

  // Stage W (128x64 row-major), repacking rows 2p/2p+1 interleaved by column.
  for (int i = t; i < 128 * 64; i += 256) {
    int k = i >> 6, n = i & 63;
    sW2[(k >> 1) * WP_STRIDE + (n << 1) + (k & 1)] = W[i];
  }
  // Stage A rows: [x_row | nsum_row] per node, float4 granularity.
  for (int i = t; i < 32 * 32; i += 256) {
    int r = i >> 5;
    int c4 = (i & 31) << 2;                       // 0..124
    long long node = nodeBase + r;
    const float* srcp = (c4 < DIM) ? (x + node * DIM + c4)
                                   : (nsum + node * DIM + (c4 - DIM));
    float4 v = *(const float4*)srcp;
    float* dp = &sA[r * 130 + c4];
    dp[0] = v.x; dp[1] = v.y; dp[2] = v.z; dp[3] = v.w;
  }
  __syncthreads();

  const int lane = t & 31;
  const int wave = t >> 5;
  const int baseRow = (wave >> 2) * 16;   // 0 or 16
  const int nTile   = (wave & 3) * 16;    // 0,16,32,48
  const int m  = lane & 15;               // A: M = lane&15
  const int kb = (lane >> 4) * 2;         // A/B: K sub-offset {0,2}
  const int n  = nTile + (lane & 15);     // B/C: N = lane&15

  v8f acc = {};
  const float* aRow  = &sA[(baseRow + m) * 130];
  // B fragment base: pair index p = kk*2 + (lane>>4); element = 2n + {0,1}.
  const float* bBase = &sW2[(lane >> 4) * WP_STRIDE + (n << 1)];
  #pragma unroll
  for (int kk = 0; kk < 32; ++kk) {
    int k = kk * 4 + kb;
    v2f a; a.x = aRow[k]; a.y = aRow[k + 1];
    v2f b = *(const v2f*)(bBase + kk * (2 * WP_STRIDE));  // one ds_load_b64
    acc = __builtin_amdgcn_wmma_f32_16x16x4_f32(
        /*neg_a=*/false, a, /*neg_b=*/false, b,
        /*c_mod=*/(short)0, acc, /*reuse_a=*/false, /*reuse_b=*/false);
  }

  // Bias-add and spill the 16x16 tile to LDS for the cross-wave LayerNorm.
  const float bv = bias[n];
  #pragma unroll
  for (int r = 0; r < 8; ++r) {
    int row = baseRow + r + (lane >> 4) * 8;   // C layout: VGPR r -> M=r / r+8
    sH[row * 68 + n] = acc[r] + bv;
  }
  __syncthreads();

  // Per-row mean / rsigma (32 rows, one thread each).
  if (t < 32) {
    float s = 0.f, s2 = 0.f;
    #pragma unroll
    for (int j = 0; j < 64; ++j) {
      float v = sH[t * 68 + j];
      s += v; s2 += v * v;
    }
    float mu  = s  * (1.f / 64.f);
    float var = s2 * (1.f / 64.f) - mu * mu;
    sStat[t]      = mu;
    sStat[32 + t] = rsqrtf(var + EPS_LN);
  }
  __syncthreads();

  // Normalize + affine + store: 8 outputs per thread.
  const int row = t >> 3;
  const int j0  = (t & 7) << 3;
  const float mu = sStat[row];
  const float rs = sStat[32 + row];
  const long long node = nodeBase + row;
  #pragma unroll
  for (int j = 0; j < 8; ++j) {
    int jj = j0 + j;
    out[node * OUTF + jj] = (sH[row * 68 + jj] - mu) * rs * gamma[jj] + beta[jj];
  }
}

// ---------------------------------------------------------------------------
extern "C" void kernel_launch(void* const* d_in, const int* in_sizes, int n_in,
                              void* d_out, int out_size, void* d_ws, size_t ws_size,
                              hipStream_t stream) {
  const float*     x     = (const float*)d_in[0];
  const long long* ei    = (const long long*)d_in[1];  // int64 edge_index [2][E]
  const float*     W     = (const float*)d_in[2];      // [128][64]
  const float*     b     = (const float*)d_in[3];      // [64]
  const float*     gamma = (const float*)d_in[4];
  const float*     beta  = (const float*)d_in[5];
  float*           out   = (float*)d_out;
  float*           nsum  = (float*)d_ws;               // [N_NODES][64] f32

  const int n4 = N_NODES * DIM / 4;
  dgc_zero_kernel<<<(n4 + 255) / 256, 256, 0, stream>>>((float4*)nsum, n4);

  const unsigned total = (unsigned)N_EDGES * 16u;
  dgc_scatter_kernel<<<(total + 255) / 256, 256, 0, stream>>>(x, ei, nsum);

  dgc_gemm_ln_kernel<<<N_NODES / 32, 256, 0, stream>>>(x, nsum, W, b, gamma,
                                                       beta, out);
}